// RegularizationLoss_5583457484972
// MI455X (gfx1250) — compile-verified
//
#include <hip/hip_runtime.h>
#include <hip/hip_bf16.h>

// RegularizationLoss: sum((dx^2 + dy^2 + dz^2)) / (B*C) over forward diffs of a
// (4,3,128,128,128) f32 mesh. Memory-bound: ~100.7MB unique reads -> ~4.3us at
// 23.3 TB/s. MI455X's 192MB L2 holds the whole mesh, so the 3x logical row
// reads (center, x+1, y+1) only cost HBM once.

typedef __attribute__((ext_vector_type(2))) float v2f;
typedef __attribute__((ext_vector_type(8))) float v8f;

#define RL_NBLK 4096
#define RL_TPB  256

// Geometry constants
#define RL_N    12            // B*C
#define RL_D    128
#define RL_I    127           // interior extent per spatial dim
#define RL_ROWS (RL_N * RL_I * RL_I)   // one row = 128 floats along z for fixed (n,x,y)

__global__ __launch_bounds__(RL_TPB)
void RegularizationLoss_main(const float* __restrict__ mesh,
                             float* __restrict__ partial) {
    const int w    = threadIdx.x >> 5;   // wave id within block (8 waves)
    const int lane = threadIdx.x & 31;
    const int rowGroups = (RL_ROWS + 7) / 8;

    float s = 0.0f;

    for (int rg = (int)blockIdx.x; rg < rowGroups; rg += (int)gridDim.x) {
        const int r = rg * 8 + w;        // row index: (n, x, y)
        if (r < RL_ROWS) {
            const int n   = r / (RL_I * RL_I);
            const int rem = r - n * (RL_I * RL_I);
            const int x   = rem / RL_I;
            const int y   = rem - x * RL_I;
            const int base = ((n * RL_D + x) * RL_D + y) * RL_D;  // max ~25.1M, fits int

            // One wave covers the whole z-row: lane handles z = 4*lane .. 4*lane+3.
            const float4* pc = reinterpret_cast<const float4*>(mesh + base);
            const float4* px = reinterpret_cast<const float4*>(mesh + base + RL_D * RL_D); // x+1
            const float4* py = reinterpret_cast<const float4*>(mesh + base + RL_D);        // y+1
            const float4 c  = pc[lane];
            const float4 cx = px[lane];
            const float4 cy = py[lane];
            // z+1 neighbor of this lane's last element lives in lane+1's .x
            const float cnext = __shfl_down(c.x, 1, 32);

            const float cv[4] = { c.x,  c.y,  c.z,  c.w  };
            const float xv[4] = { cx.x, cx.y, cx.z, cx.w };
            const float yv[4] = { cy.x, cy.y, cy.z, cy.w };
            const float zv[4] = { c.y,  c.z,  c.w,  cnext };

#pragma unroll
            for (int j = 0; j < 4; ++j) {
                const int z = 4 * lane + j;
                if (z < RL_I) {          // only lane31/j==3 (z==127) is excluded
                    const float a  = cv[j];
                    const float dx = xv[j] - a;
                    const float dy = yv[j] - a;
                    const float dz = zv[j] - a;
                    s = fmaf(dx, dx, s);
                    s = fmaf(dy, dy, s);
                    s = fmaf(dz, dz, s);
                }
            }
        }
    }

    // Deterministic intra-wave tree reduction (wave32)
#pragma unroll
    for (int off = 16; off > 0; off >>= 1)
        s += __shfl_down(s, off, 32);

    __shared__ float wsum[8];
    if (lane == 0) wsum[w] = s;
    __syncthreads();
    if (threadIdx.x == 0) {
        float bs = 0.0f;
#pragma unroll
        for (int i = 0; i < 8; ++i) bs += wsum[i];   // fixed order -> deterministic
        partial[blockIdx.x] = bs;
    }
}

// Single-wave final reduction. The 32 lane partials are summed with
// V_WMMA_F32_16X16X4_F32: A(16x4) holds lane values (lane L -> M=L%16,
// K=0 for L<16 / K=2 for L>=16 in VGPR0; VGPR1 = 0), B = ones(4x16).
// D[m][n] = p[m] + p[m+16]; summing the 8 D VGPRs per lane gives
// t(lane<16) = sum_{m=0..7} D[m], t(lane>=16) = sum_{m=8..15} D[m],
// so total = t(lane0) + t(lane16).
__global__ void RegularizationLoss_final(const float* __restrict__ partial,
                                         float* __restrict__ out,
                                         int count) {
    const int lane = threadIdx.x;   // blockDim.x == 32
    float p = 0.0f;
    for (int i = lane; i < count; i += 32)   // fixed order per lane -> deterministic
        p += partial[i];

    v2f a; a.x = p;    a.y = 0.0f;
    v2f b; b.x = 1.0f; b.y = 1.0f;
    v8f acc = {};
    acc = __builtin_amdgcn_wmma_f32_16x16x4_f32(
        /*neg_a=*/false, a, /*neg_b=*/false, b,
        /*c_mod=*/(short)0, acc, /*reuse_a=*/false, /*reuse_b=*/false);

    float t = acc[0] + acc[1] + acc[2] + acc[3] + acc[4] + acc[5] + acc[6] + acc[7];
    t += __shfl_down(t, 16, 32);

    if (lane == 0) out[0] = t * (1.0f / (float)RL_N);
}

extern "C" void kernel_launch(void* const* d_in, const int* in_sizes, int n_in,
                              void* d_out, int out_size, void* d_ws, size_t ws_size,
                              hipStream_t stream) {
    const float* mesh = (const float*)d_in[0];
    float* partial    = (float*)d_ws;
    float* out        = (float*)d_out;

    // One float partial per block in d_ws; cap (to a multiple of 32) if the
    // workspace is unexpectedly small. ws_size is fixed -> deterministic.
    int nblk = RL_NBLK;
    const int maxPartials = (int)((ws_size / sizeof(float)) & ~(size_t)31);
    if (nblk > maxPartials) nblk = maxPartials > 32 ? maxPartials : 32;

    RegularizationLoss_main<<<nblk, RL_TPB, 0, stream>>>(mesh, partial);
    RegularizationLoss_final<<<1, 32, 0, stream>>>(partial, out, nblk);
}